// torch_shift_82489141887540
// MI455X (gfx1250) — compile-verified
//
#include <hip/hip_runtime.h>
#include <cstdint>

// ----------------------------------------------------------------------------
// torch_shift (dim=3) on MI455X / gfx1250.
//
// Pure permutation-with-zero-fill:
//   out[b, co, d, hp, w] = x[b, co/4, d + 2 - j, wrap(hp*4 + co%4), w]
//   with j = (co>>3) % 5; zero when (d + 2 - j) outside [0,64);
//   wrap(h) = h - 66 when h >= 66 (circular H pad by 2).
//
// Bandwidth-bound: 173 MB in + 178 MB out -> ~15 us at 23.3 TB/s.
// CDNA5 path exercised: async data mover (global_load_async_to_lds_b128 /
// global_store_async_from_lds_b128) + split s_wait_asynccnt counter.
// ----------------------------------------------------------------------------

namespace {
constexpr int Bn = 4, Cn = 40, Dn = 64, Hn = 66, Wn = 64;
constexpr int SHIFTS = 5;
constexpr int HP = 17;                 // (66 + 2) / 4
constexpr int CO = Cn * 4;             // 160
constexpr int OUT_ELEMS = Bn * CO * Dn * HP * Wn;   // 44,564,480
constexpr int CHUNKS = OUT_ELEMS / 4;               // 11,141,120 float4 chunks
constexpr int TPB = 256;
constexpr int CPT = 4;                               // chunks per thread
constexpr int CHUNKS_PER_BLOCK = TPB * CPT;          // 1024
constexpr int GRID = CHUNKS / CHUNKS_PER_BLOCK;      // 10,880 (exact)
}  // namespace

__global__ __launch_bounds__(TPB) void shift3d_async_kernel(
    const float* __restrict__ x, float* __restrict__ out, int out_size) {
  // Staging buffer for async global->LDS->global streaming: one 16B slot per
  // (thread, batch-slot). 16 KB of the WGP's 320 KB LDS.
  __shared__ float4 stage[CPT * TPB];

  const uint32_t tid = threadIdx.x;
  // Low 32 bits of a generic pointer to LDS == workgroup-relative LDS byte
  // offset (flat-address aperture layout), which is what the async ops take.
  const uint32_t lds_base = (uint32_t)(uintptr_t)(&stage[0]);

  const uint64_t src_base = (uint64_t)(uintptr_t)x;
  const uint64_t dst_base = (uint64_t)(uintptr_t)out;

  const uint32_t block_chunk = (uint32_t)blockIdx.x * CHUNKS_PER_BLOCK;

  uint64_t dst_addr[CPT];
  uint32_t lds_addr[CPT];
  bool valid[CPT];

  // ---- Batch of CPT async loads (consecutive lanes -> consecutive 16B) ----
#pragma unroll
  for (int k = 0; k < CPT; ++k) {
    const uint32_t cidx = block_chunk + (uint32_t)k * TPB + tid;

    const uint32_t w4 = cidx & 15;       // float4 index within the W=64 row
    const uint32_t row = cidx >> 4;      // (b, co, d, hp)
    const uint32_t hp = row % HP;
    const uint32_t t = row / HP;
    const uint32_t d = t & 63;
    const uint32_t t2 = t >> 6;
    const uint32_t co = t2 % CO;
    const uint32_t b = t2 / CO;

    const uint32_t c = co >> 2;                    // source channel
    const uint32_t p = co & 3;                     // H-fold sub-row
    const uint32_t j = (co >> 3) % SHIFTS;         // shift group
    const int32_t dsrc = (int32_t)d + 2 - (int32_t)j;
    uint32_t h = hp * 4 + p;
    if (h >= (uint32_t)Hn) h -= (uint32_t)Hn;      // circular H pad wrap

    const uint32_t lds = lds_base + ((uint32_t)k * TPB + tid) * 16u;
    const uint64_t dst = dst_base + (uint64_t)cidx * 16u;

    dst_addr[k] = dst;
    lds_addr[k] = lds;
    const bool ok = (dsrc >= 0) & (dsrc < Dn);
    valid[k] = ok;

    if (ok) {
      const uint64_t src_elem =
          ((((uint64_t)(b * Cn + c) * Dn + (uint32_t)dsrc) * Hn + h) * Wn +
           (uint64_t)w4 * 4u);
      const uint64_t src = src_base + src_elem * 4u;
      // CDNA5 async data mover: 16B global -> LDS per lane, ASYNCcnt-tracked.
      asm volatile("global_load_async_to_lds_b128 %0, %1, off"
                   :
                   : "v"(lds), "v"(src)
                   : "memory");
    } else {
      // Depth-shift out of range -> zero fill directly (skip LDS round trip).
      float4 z;
      z.x = 0.f; z.y = 0.f; z.z = 0.f; z.w = 0.f;
      *(float4*)(uintptr_t)dst = z;
    }
  }

  // Wait for the whole load batch to land in LDS.
  asm volatile("s_wait_asynccnt 0" ::: "memory");

  // ---- Batch of CPT async stores: LDS -> global, 16B per lane ----
#pragma unroll
  for (int k = 0; k < CPT; ++k) {
    if (valid[k]) {
      asm volatile("global_store_async_from_lds_b128 %0, %1, off"
                   :
                   : "v"(dst_addr[k]), "v"(lds_addr[k])
                   : "memory");
    }
  }

  // Second output of the reference tuple: pad_1 == 2 (scalar), appended flat.
  if (blockIdx.x == 0 && tid == 0 && out_size > OUT_ELEMS) {
    out[OUT_ELEMS] = 2.0f;
  }

  // Drain async stores (S_ENDPGM also implies wait-idle; explicit for safety).
  asm volatile("s_wait_asynccnt 0" ::: "memory");
}

extern "C" void kernel_launch(void* const* d_in, const int* in_sizes, int n_in,
                              void* d_out, int out_size, void* d_ws,
                              size_t ws_size, hipStream_t stream) {
  (void)in_sizes; (void)n_in; (void)d_ws; (void)ws_size;
  const float* x = (const float*)d_in[0];
  float* out = (float*)d_out;
  shift3d_async_kernel<<<GRID, TPB, 0, stream>>>(x, out, out_size);
}